// TimeDecayLoss_72395968741464
// MI455X (gfx1250) — compile-verified
//
#include <hip/hip_runtime.h>

// Problem constants (from reference)
#define HH 512
#define WWID 512
#define CC 2048        // 2H + 2W
#define BB 4
#define SS 4096
#define TEMP_INV (1.0f / 256.0f)

typedef float v2f __attribute__((ext_vector_type(2)));
typedef float v4f __attribute__((ext_vector_type(4)));
typedef float v8f __attribute__((ext_vector_type(8)));

// ---------------------------------------------------------------------------
// Kernel 1: per-16-block cumulative decay products.
//   d[s]   = exp(-(t[s+1]-t[s])/TEMP), d[S-1]=0
//   cum[b][blk*16+i] = prod_{k<i} d[blk*16+k]          (cum at i=0 is 1)
//   blockprod[b][blk] = prod over the whole 16-block
// ---------------------------------------------------------------------------
__global__ void __launch_bounds__(256)
prep_kernel(const float* __restrict__ target,
            float* __restrict__ cum, float* __restrict__ blockprod) {
  const int nblk = SS / 16;
  int gid = blockIdx.x * blockDim.x + threadIdx.x;
  if (gid >= BB * nblk) return;
  int b = gid / nblk;
  int blk = gid - b * nblk;
  int s0 = blk * 16;
  const float* tg = target + (size_t)b * SS * 4;
  float t_cur = tg[(size_t)s0 * 4];
  float prod = 1.0f;
#pragma unroll
  for (int i = 0; i < 16; ++i) {
    int s = s0 + i;
    cum[(size_t)b * SS + s] = prod;
    float d;
    if (s < SS - 1) {
      float t_next = tg[(size_t)(s + 1) * 4];
      d = __expf(-(t_next - t_cur) * TEMP_INV);
      t_cur = t_next;
    } else {
      d = 0.0f;   // reference concatenates a zero decay at the end
    }
    prod *= d;
  }
  blockprod[(size_t)b * nblk + blk] = prod;
}

// ---------------------------------------------------------------------------
// Kernel 2: blocked reverse scan via V_WMMA_F32_16X16X4_F32.
//   Per 16-step block:  r[16 x 16ch] = W(16x16 upper-tri) @ a_block(16x16ch)
//                                      + Q[i] (x) carry            (K-slice 5)
//   W[i,j] = cum[j]/cum[i] (j>=i), Q[i] = blockprod/cum[i].
// One single-wave workgroup = one b x 32 channels: TWO independent 16-channel
// WMMA chains share the A (W) operands and the target-row loads, giving 2-way
// ILP on the matrix pipe.  Branchless sparse-a gather (no exec ladders).
// ---------------------------------------------------------------------------
__global__ void __launch_bounds__(32)
scan_kernel(const float* __restrict__ target,
            const float* __restrict__ cum,
            const float* __restrict__ blockprod,
            float* __restrict__ r) {
  const int nPair = CC / 32;                   // 64 channel-pairs per b
  const int b     = blockIdx.x / nPair;
  const int pair  = blockIdx.x - b * nPair;
  const int lane  = threadIdx.x & 31;
  const int half  = lane >> 4;                 // 0 / 1
  const int li    = lane & 15;                 // 0..15
  const int c0    = pair * 32;                 // 32-channel window (one seg)
  const int cA    = c0 + li;                   // tile A column N
  const int seg   = c0 >> 9;                   // 512-wide segment id
  const int use_w = seg >> 1;                  // key = w (1) or h (0)
  const int use_p = seg & 1;                   // val = p (1) or 1-p (0)
  const int clA   = cA & 511;                  // local channel in segment
  const int clB   = clA + 16;

  const v4f*  tgrow = (const v4f*)(target + (size_t)b * SS * 4);
  const float* cumb = cum       + (size_t)b * SS;
  const float* bp   = blockprod + (size_t)b * (SS / 16);
  float* rbA = r + (size_t)b * SS * CC + cA;   // tile A column base
  float* rbB = rbA + 16;                       // tile B column base

  float carryA = 0.0f, carryB = 0.0f;          // r[s_blk+16, col] (lanes 0-15)
  for (int sblk = SS - 16; sblk >= 0; sblk -= 16) {
    const float rcpi  = __builtin_amdgcn_rcpf(cumb[sblk + li]); // 1/cum[i]
    const float cpblk = bp[sblk >> 4];

    // ---- A operands (shared by both tiles) + branchless target gather.
    //      A 16x4 layout: VGPR0 -> K = 0|2 (by half), VGPR1 -> K = 1|3.
    v2f a_op[5];
    float keyx[4], keyy[4], valx[4], valy[4];
#pragma unroll
    for (int k = 0; k < 4; ++k) {
      int jx = 4 * k + 2 * half;
      int jy = jx + 1;
      float cx = cumb[sblk + jx];
      float cy = cumb[sblk + jy];
      a_op[k].x = (jx >= li) ? cx * rcpi : 0.0f;
      a_op[k].y = (jy >= li) ? cy * rcpi : 0.0f;
      v4f rx = tgrow[sblk + jx];               // {t, h, w, p} b128 load
      v4f ry = tgrow[sblk + jy];
      keyx[k] = use_w ? rx[2] : rx[1];
      valx[k] = use_p ? rx[3] : (1.0f - rx[3]);
      keyy[k] = use_w ? ry[2] : ry[1];
      valy[k] = use_p ? ry[3] : (1.0f - ry[3]);
    }
    a_op[4].x = half ? 0.0f : cpblk * rcpi;    // column 16 = Q[i]
    a_op[4].y = 0.0f;                          // columns 17..19 = 0

    // ---- B operands for both tiles (B 4x16 layout mirrors A; N = lane%16).
    v2f bA[5], bB[5];
#pragma unroll
    for (int k = 0; k < 4; ++k) {
      int kx = (int)keyx[k], ky = (int)keyy[k];
      bA[k].x = (kx == clA) ? valx[k] : 0.0f;
      bA[k].y = (ky == clA) ? valy[k] : 0.0f;
      bB[k].x = (kx == clB) ? valx[k] : 0.0f;
      bB[k].y = (ky == clB) ? valy[k] : 0.0f;
    }
    bA[4].x = half ? 0.0f : carryA;  bA[4].y = 0.0f;   // K-slice 5: carry
    bB[4].x = half ? 0.0f : carryB;  bB[4].y = 0.0f;

    // ---- Two interleaved accumulating fp32 WMMA chains (independent).
    v8f accA = {}, accB = {};
#pragma unroll
    for (int k = 0; k < 5; ++k) {
      accA = __builtin_amdgcn_wmma_f32_16x16x4_f32(false, a_op[k], false, bA[k],
                                                   (short)0, accA, false, false);
      accB = __builtin_amdgcn_wmma_f32_16x16x4_f32(false, a_op[k], false, bB[k],
                                                   (short)0, accB, false, false);
    }

    // New carries = r[s_blk] = row M=0 -> acc[0] on lanes 0..15 (gated above).
    carryA = accA[0];
    carryB = accB[0];

    // ---- Store: C/D layout: VGPR k -> M = k + 8*half, N = lane%16.
#pragma unroll
    for (int k = 0; k < 8; ++k) {
      size_t row = (size_t)(sblk + k + 8 * half) * CC;
      rbA[row] = accA[k];
      rbB[row] = accB[k];
    }
  }

  // Reference overwrites r[:,0,:] = a[:,0,:].
  if (half == 0) {
    v4f r0 = tgrow[0];
    float key = use_w ? r0[2] : r0[1];
    float val = use_p ? r0[3] : (1.0f - r0[3]);
    rbA[0] = ((int)key == clA) ? val : 0.0f;
    rbB[0] = ((int)key == clB) ? val : 0.0f;
  }
}

// ---------------------------------------------------------------------------
// Kernel 3: fused soft-CE per (b,s).  4 waves, one 512-wide segment each.
//   CE = mp + log(sum e^{pred-mp}) - (sum e^{r-mr} * pred) / (sum e^{r-mr})
// pred is read ONCE -> non-temporal loads keep the 134MB r resident in the
// 192MB L2 (written by the scan, re-read here at L2 bandwidth).
// ---------------------------------------------------------------------------
__global__ void __launch_bounds__(128)
ce_kernel(const float* __restrict__ pred,
          const float* __restrict__ target,
          const float* __restrict__ r,
          float* __restrict__ partials) {
  const int bs   = blockIdx.x;                 // 0 .. B*S-1
  const int b    = bs / SS;
  const int s    = bs - b * SS;
  const int wave = threadIdx.x >> 5;           // segment q = 0..3
  const int lane = threadIdx.x & 31;
  const size_t off = ((size_t)b * SS + s) * CC + (size_t)wave * 512;
  const float* pr = pred + off;
  const float* rr = r + off;

  v4f pv[4], rv[4];
#pragma unroll
  for (int k = 0; k < 4; ++k) {
    int idx = k * 128 + lane * 4;
    pv[k] = __builtin_nontemporal_load((const v4f*)(pr + idx)); // TH_NT
    rv[k] = *(const v4f*)(rr + idx);                            // RT: L2 hit
  }

  float mp = -INFINITY, mr = -INFINITY;
#pragma unroll
  for (int k = 0; k < 4; ++k) {
#pragma unroll
    for (int j = 0; j < 4; ++j) {
      mp = fmaxf(mp, pv[k][j]);
      mr = fmaxf(mr, rv[k][j]);
    }
  }
#pragma unroll
  for (int d = 16; d >= 1; d >>= 1) {
    mp = fmaxf(mp, __shfl_xor(mp, d));
    mr = fmaxf(mr, __shfl_xor(mr, d));
  }

  float zp = 0.0f, zr = 0.0f, t = 0.0f;
#pragma unroll
  for (int k = 0; k < 4; ++k) {
#pragma unroll
    for (int j = 0; j < 4; ++j) {
      float e = __expf(rv[k][j] - mr);
      zr += e;
      t  += e * pv[k][j];
      zp += __expf(pv[k][j] - mp);
    }
  }
#pragma unroll
  for (int d = 16; d >= 1; d >>= 1) {
    zp += __shfl_xor(zp, d);
    zr += __shfl_xor(zr, d);
    t  += __shfl_xor(t, d);
  }

  __shared__ float sdata[4];
  if (lane == 0) {
    float ce = mp + __logf(zp) - t / zr;
    float p  = target[((size_t)b * SS + s) * 4 + 3];
    float wgt = (wave & 1) ? p : (1.0f - p);   // q=1,3 -> p ; q=0,2 -> 1-p
    sdata[wave] = ce * wgt;
  }
  __syncthreads();
  if (threadIdx.x == 0) {
    partials[bs] = (sdata[0] + sdata[1] + sdata[2] + sdata[3]) *
                   (1.0f / (float)(BB * SS));
  }
}

// ---------------------------------------------------------------------------
// Kernel 4: deterministic tree reduction of B*S partials -> scalar loss.
// ---------------------------------------------------------------------------
__global__ void __launch_bounds__(256)
reduce_kernel(const float* __restrict__ partials, float* __restrict__ out) {
  __shared__ float sm[256];
  float acc = 0.0f;
  for (int i = threadIdx.x; i < BB * SS; i += 256) acc += partials[i];
  sm[threadIdx.x] = acc;
  __syncthreads();
#pragma unroll
  for (int offt = 128; offt >= 1; offt >>= 1) {
    if ((int)threadIdx.x < offt) sm[threadIdx.x] += sm[threadIdx.x + offt];
    __syncthreads();
  }
  if (threadIdx.x == 0) out[0] = sm[0];
}

// ---------------------------------------------------------------------------
extern "C" void kernel_launch(void* const* d_in, const int* in_sizes, int n_in,
                              void* d_out, int out_size, void* d_ws, size_t ws_size,
                              hipStream_t stream) {
  (void)in_sizes; (void)n_in; (void)out_size; (void)ws_size;
  const float* pred   = (const float*)d_in[0];   // (B,S,C) fp32
  const float* target = (const float*)d_in[1];   // (B,S,4) fp32
  float* out = (float*)d_out;                    // scalar loss

  // Workspace partition
  float* r         = (float*)d_ws;                         // B*S*C floats
  float* cum       = r + (size_t)BB * SS * CC;             // B*S
  float* blockprod = cum + (size_t)BB * SS;                // B*(S/16)
  float* partials  = blockprod + (size_t)BB * (SS / 16);   // B*S

  // 1) decay cumulative products per 16-block
  {
    int n = BB * (SS / 16);
    prep_kernel<<<(n + 255) / 256, 256, 0, stream>>>(target, cum, blockprod);
  }
  // 2) blocked reverse scan on the WMMA pipe (writes full r), 256 single-wave
  //    workgroups, each running two independent 5-WMMA chains per step.
  scan_kernel<<<BB * (CC / 32), 32, 0, stream>>>(target, cum, blockprod, r);
  // 3) fused soft-CE per (b,s)
  ce_kernel<<<BB * SS, 128, 0, stream>>>(pred, target, r, partials);
  // 4) deterministic final reduction
  reduce_kernel<<<1, 256, 0, stream>>>(partials, out);
}